// NormAdd_58737972740836
// MI455X (gfx1250) — compile-verified
//
#include <hip/hip_runtime.h>
#include <hip/hip_bf16.h>
#include <stdint.h>

// out = relu( x0 + x1 + (b0 + b1 + conv_bias) ), NHWC fp32, C = 256.
// Memory-bound (AI ~ 0.33 FLOP/B): optimize for pure HBM streaming on MI455X.

typedef float v4f __attribute__((ext_vector_type(4)));

#define THREADS 256
#define CCH 256          // channel count (bias length)
#define CCH4 (CCH / 4)   // float4 groups per channel block

#define AS1 __attribute__((address_space(1)))
#define AS3 __attribute__((address_space(3)))

__global__ __launch_bounds__(THREADS) void normadd_relu_kernel(
    const float* __restrict__ x0,
    const float* __restrict__ x1,
    const float* __restrict__ b0,
    const float* __restrict__ b1,
    const float* __restrict__ cb,
    float* __restrict__ out,
    long long n)
{
    __shared__ float s_raw[3 * CCH];   // staged b0 | b1 | conv_bias
    __shared__ float s_bsum[CCH];      // b0 + b1 + conv_bias

    const int t = threadIdx.x;

    // ---- Stage the three bias vectors into LDS ------------------------------
#if defined(__AMDGCN__) && __has_builtin(__builtin_amdgcn_global_load_async_to_lds_b32)
    // CDNA5 async DMA path: per-lane 4B copies global->LDS, tracked by ASYNCcnt.
    // Builtin signature is typed: (global int*, lds int*, imm offset, imm cpol).
    __builtin_amdgcn_global_load_async_to_lds_b32(
        (AS1 int*)(b0 + t), (AS3 int*)(s_raw + t), 0, 0);
    __builtin_amdgcn_global_load_async_to_lds_b32(
        (AS1 int*)(b1 + t), (AS3 int*)(s_raw + CCH + t), 0, 0);
    __builtin_amdgcn_global_load_async_to_lds_b32(
        (AS1 int*)(cb + t), (AS3 int*)(s_raw + 2 * CCH + t), 0, 0);
    __builtin_amdgcn_s_wait_asynccnt(0);
#else
    // Fallback: plain loads (biases are tiny and L2-resident).
    s_raw[t]            = b0[t];
    s_raw[CCH + t]      = b1[t];
    s_raw[2 * CCH + t]  = cb[t];
#endif
    __syncthreads();

    // Fold the three bias vectors once per block.
    s_bsum[t] = s_raw[t] + s_raw[CCH + t] + s_raw[2 * CCH + t];
    __syncthreads();

    // ---- Steady-state streaming loop ---------------------------------------
    // Quad index i advances by gridDim*THREADS (a multiple of 64), so the
    // channel group (i & 63) is loop-invariant per thread: hoist the bias.
    const long long n4     = n >> 2;
    const long long stride = (long long)gridDim.x * THREADS;
    long long       i      = (long long)blockIdx.x * THREADS + t;

    const v4f bs = ((const v4f*)s_bsum)[t & (CCH4 - 1)];

    const v4f* __restrict__ x04  = (const v4f*)x0;
    const v4f* __restrict__ x14  = (const v4f*)x1;
    v4f* __restrict__       out4 = (v4f*)out;

    for (; i < n4; i += stride) {
        // Streamed once -> non-temporal (th:NT) B128 traffic.
        v4f a = __builtin_nontemporal_load(x04 + i);
        v4f b = __builtin_nontemporal_load(x14 + i);
        v4f r = a + b + bs;
        r.x = fmaxf(r.x, 0.0f);
        r.y = fmaxf(r.y, 0.0f);
        r.z = fmaxf(r.z, 0.0f);
        r.w = fmaxf(r.w, 0.0f);
        __builtin_nontemporal_store(r, out4 + i);
    }

    // ---- Scalar tail (dead for the reference shape; kept for generality) ---
    const long long tail_begin = n4 << 2;
    for (long long j = tail_begin + (long long)blockIdx.x * THREADS + t;
         j < n; j += stride) {
        float r = x0[j] + x1[j] + s_bsum[j % CCH];
        out[j] = fmaxf(r, 0.0f);
    }
}

extern "C" void kernel_launch(void* const* d_in, const int* in_sizes, int n_in,
                              void* d_out, int out_size, void* d_ws, size_t ws_size,
                              hipStream_t stream) {
    const float* x0 = (const float*)d_in[0];
    const float* x1 = (const float*)d_in[1];
    const float* b0 = (const float*)d_in[2];
    const float* b1 = (const float*)d_in[3];
    const float* cb = (const float*)d_in[4];
    float*       out = (float*)d_out;

    const long long n  = (long long)in_sizes[0];
    const long long n4 = n >> 2;

    // 2048 blocks x 8 waves (wave32) -> ~16k waves in flight, ~12 quads/thread:
    // enough outstanding B128 requests to saturate 23.3 TB/s HBM.
    long long blocks = (n4 + THREADS - 1) / THREADS;
    if (blocks > 2048) blocks = 2048;
    if (blocks < 1)    blocks = 1;

    normadd_relu_kernel<<<(int)blocks, THREADS, 0, stream>>>(
        x0, x1, b0, b1, cb, out, n);
}